// SSIM_81217831567710
// MI455X (gfx1250) — compile-verified
//
#include <hip/hip_runtime.h>

typedef __attribute__((ext_vector_type(2))) float v2f;
typedef __attribute__((ext_vector_type(8))) float v8f;

#define H_IMG 512
#define W_IMG 512
#define TILE 16
#define TPITCH 30          // input tile row pitch (floats), even -> 8B aligned v2f loads
#define TROWS 32           // 26 real rows + zero pad rows for block-1 A operand
#define HPITCH 17          // horizontal-stage pitch (odd -> spread banks)
#define HPLANE (32 * HPITCH)
#define NTILES (32 * 32 * 48)

// D = A(16x4) * B(4x16) + C   (fp32 WMMA, wave32)
__device__ __forceinline__ v8f wmma4(v2f a, v2f b, v8f c) {
  return __builtin_amdgcn_wmma_f32_16x16x4_f32(false, a, false, b, (short)0, c,
                                               false, false);
}

__global__ __launch_bounds__(32) void ssim_tile_kernel(
    const float* __restrict__ img1, const float* __restrict__ img2,
    float* __restrict__ part) {
  __shared__ __align__(16) float tile1[TROWS * TPITCH];
  __shared__ __align__(16) float tile2[TROWS * TPITCH];
  __shared__ __align__(16) float hbuf[5 * HPLANE];
  __shared__ float wlds[16];

  const int lane = threadIdx.x;       // 0..31
  const int h2 = (lane >> 4) << 1;    // 2*h (half-wave select)
  const int nm = lane & 15;           // row (A) or col (B) index

  // Gaussian window, sigma = 1.5, 11 taps, normalized (matches reference)
  {
    float myg = 0.f, gsum = 0.f;
#pragma unroll
    for (int i = 0; i < 11; ++i) {
      float d = (float)(i - 5);
      float e = __expf(-(d * d) / 4.5f);
      gsum += e;
      if (i == lane) myg = e;
    }
    if (lane < 11) wlds[lane] = myg / gsum;
  }
  __syncthreads();

  const int x0 = blockIdx.x * TILE;
  const int y0 = blockIdx.y * TILE;
  const size_t pbase = (size_t)blockIdx.z * (H_IMG * W_IMG);

  // Load zero-padded 26x26 halo region (stored in 32x30 zeroed tile)
  for (int idx = lane; idx < TROWS * TPITCH; idx += 32) {
    int r = idx / TPITCH;
    int c = idx - r * TPITCH;
    int y = y0 - 5 + r;
    int x = x0 - 5 + c;
    float a = 0.f, b = 0.f;
    if (r < 26 && c < 26 && y >= 0 && y < H_IMG && x >= 0 && x < W_IMG) {
      size_t off = pbase + (size_t)y * W_IMG + x;
      a = img1[off];
      b = img2[off];
    }
    tile1[idx] = a;
    tile2[idx] = b;
  }
  __syncthreads();

  // Banded Toeplitz fragments: value(k, nm) = w[k - nm] if 0<=k-nm<=10.
  // Same per-lane layout serves as horizontal-B (k=row of B, nm=col) and
  // vertical-A (k=K index, nm=row).  k = 4*s + j + 2h, j in {0,1}.
  v2f wfrag[7];
#pragma unroll
  for (int s = 0; s < 7; ++s) {
    int d0 = (4 * s + h2) - nm;
    int d1 = d0 + 1;
    wfrag[s].x = (d0 >= 0 && d0 <= 10) ? wlds[d0] : 0.f;
    wfrag[s].y = (d1 >= 0 && d1 <= 10) ? wlds[d1] : 0.f;
  }

  const v8f zero8 = {0.f, 0.f, 0.f, 0.f, 0.f, 0.f, 0.f, 0.f};

  // ---------------- Horizontal pass: Hout[r][n] = sum_k w[k]*tile[r][n+k]
  // Two 16-row blocks; 5 quantities {x1, x2, x1^2, x2^2, x1*x2}
#pragma unroll
  for (int b = 0; b < 2; ++b) {
    v8f hacc[5];
#pragma unroll
    for (int q = 0; q < 5; ++q) hacc[q] = zero8;
    const int rowbase = b * 16;
    const float* t1r = &tile1[(rowbase + nm) * TPITCH];
    const float* t2r = &tile2[(rowbase + nm) * TPITCH];
#pragma unroll
    for (int s = 0; s < 7; ++s) {
      const int k0 = 4 * s + h2;  // even -> 8B aligned
      v2f a = *(const v2f*)(t1r + k0);
      v2f c = *(const v2f*)(t2r + k0);
      hacc[0] = wmma4(a, wfrag[s], hacc[0]);
      hacc[1] = wmma4(c, wfrag[s], hacc[1]);
      hacc[2] = wmma4(a * a, wfrag[s], hacc[2]);
      hacc[3] = wmma4(c * c, wfrag[s], hacc[3]);
      hacc[4] = wmma4(a * c, wfrag[s], hacc[4]);
    }
    // D layout: VGPR j holds rows {j, j+8}, cols = lane&15
#pragma unroll
    for (int q = 0; q < 5; ++q) {
#pragma unroll
      for (int j = 0; j < 8; ++j) {
        int row = rowbase + j + 4 * h2;  // + 8h
        hbuf[q * HPLANE + row * HPITCH + nm] = hacc[q][j];
      }
    }
  }
  __syncthreads();

  // ---------------- Vertical pass: Out[m][n] = sum_k w[k]*Hout[m+k][n]
  v8f vacc[5];
#pragma unroll
  for (int q = 0; q < 5; ++q) vacc[q] = zero8;
#pragma unroll
  for (int s = 0; s < 7; ++s) {
    const int k0 = 4 * s + h2;
#pragma unroll
    for (int q = 0; q < 5; ++q) {
      v2f bf;
      bf.x = hbuf[q * HPLANE + k0 * HPITCH + nm];
      bf.y = hbuf[q * HPLANE + (k0 + 1) * HPITCH + nm];
      vacc[q] = wmma4(wfrag[s], bf, vacc[q]);
    }
  }

  // ---------------- SSIM map + wave reduction
  const float C1c = 0.0001f;  // 0.01^2
  const float C2c = 0.0009f;  // 0.03^2
  float ssum = 0.f;
#pragma unroll
  for (int j = 0; j < 8; ++j) {
    float mu1 = vacc[0][j], mu2 = vacc[1][j];
    float mu1sq = mu1 * mu1, mu2sq = mu2 * mu2, mu12 = mu1 * mu2;
    float s1 = vacc[2][j] - mu1sq;
    float s2 = vacc[3][j] - mu2sq;
    float s12 = vacc[4][j] - mu12;
    float num = (2.f * mu12 + C1c) * (2.f * s12 + C2c);
    float den = (mu1sq + mu2sq + C1c) * (s1 + s2 + C2c);
    ssum += num / den;
  }
#pragma unroll
  for (int off = 16; off > 0; off >>= 1) ssum += __shfl_xor(ssum, off, 32);
  if (lane == 0) {
    int tid = (blockIdx.z * gridDim.y + blockIdx.y) * gridDim.x + blockIdx.x;
    part[tid] = ssum;
  }
}

// Deterministic final reduction (fixed order, double accumulation)
__global__ __launch_bounds__(256) void ssim_reduce_kernel(
    const float* __restrict__ part, float* __restrict__ out, int n,
    double inv_count) {
  __shared__ double sbuf[256];
  double s = 0.0;
  for (int i = threadIdx.x; i < n; i += 256) s += (double)part[i];
  sbuf[threadIdx.x] = s;
  __syncthreads();
  for (int st = 128; st > 0; st >>= 1) {
    if ((int)threadIdx.x < st) sbuf[threadIdx.x] += sbuf[threadIdx.x + st];
    __syncthreads();
  }
  if (threadIdx.x == 0) out[0] = (float)(sbuf[0] * inv_count);
}

extern "C" void kernel_launch(void* const* d_in, const int* in_sizes, int n_in,
                              void* d_out, int out_size, void* d_ws,
                              size_t ws_size, hipStream_t stream) {
  (void)in_sizes;
  (void)n_in;
  (void)out_size;
  (void)ws_size;
  const float* img1 = (const float*)d_in[0];
  const float* img2 = (const float*)d_in[1];
  float* out = (float*)d_out;
  float* part = (float*)d_ws;  // 49152 floats = 192 KiB

  dim3 grid(32, 32, 48);  // 16x16 tiles over 512x512, 16*3 planes
  dim3 block(32);         // one wave32 per tile
  ssim_tile_kernel<<<grid, block, 0, stream>>>(img1, img2, part);

  const double inv_count = 1.0 / (16.0 * 3.0 * 512.0 * 512.0);
  ssim_reduce_kernel<<<1, 256, 0, stream>>>(part, out, NTILES, inv_count);
}